// LSTMDecWithAttention_13872744366343
// MI455X (gfx1250) — compile-verified
//
#include <hip/hip_runtime.h>
#include <hip/hip_bf16.h>
#include <math.h>

typedef float v2f __attribute__((ext_vector_type(2)));
typedef float v4f __attribute__((ext_vector_type(4)));
typedef float v8f __attribute__((ext_vector_type(8)));

#define HDIM 1024
#define EDIM 1024
#define LDIM 512
#define VDIM 50257

// workspace offsets (floats)
#define WS_ATTN_IN   0        // 2048 : [embedded | h]
#define WS_SCORES    2048     // 512  : attn scores -> softmax weights (in place)
#define WS_ATTN_APP  2560     // 1024 : attention-applied context
#define WS_COMB_IN   3584     // 2048 : [embedded | attn_applied]
#define WS_XVEC      5632     // 1024 : tanh(comb) = LSTM input
#define WS_GATES     6656     // 4096 : i,f,g,o pre-activations
#define WS_HNEW      10752    // 1024 : new hidden state
#define WS_LSE       11776    // 1    : logsumexp of logits

// ---------------------------------------------------------------------------
// Wave-level GEMV via V_WMMA_F32_16X16X4_F32, b128 loads, 2-stage pipeline.
//
// One wave computes a 16-row tile of y = W(rows x cols) @ x + bias [+ y].
// K-slot permutation trick: WMMA reduces over all 4 K slots, so we may map
// any 4 logical columns to the slots as long as A and B agree. We map
//   half-0 lanes (0-15):  logical cols {k, k+1}   (wmma0), {k+2, k+3} (wmma1)
//   half-1 lanes (16-31): logical cols {k+4, k+5} (wmma0), {k+6, k+7} (wmma1)
// so each lane's data for a 2-WMMA group is ONE contiguous float4:
//   W[row, k + 4*half .. k + 4*half + 3]   (global_load_b128)
// B is the x chunk broadcast across all 16 N columns with the same K map.
// After the K loop every D column equals y: lane 0 VGPR r -> row r (0..7),
// lane 16 VGPR r -> row 8+r.
// cols must be a multiple of 16 (holds: 1024 / 2048).
// ---------------------------------------------------------------------------
__global__ void wmma_gemv_kernel(const float* __restrict__ W,
                                 const float* __restrict__ x,
                                 const float* __restrict__ bias,
                                 float* __restrict__ y,
                                 int rows, int cols, int accumulate)
{
    const int lane = threadIdx.x & 31;
    const int wave = threadIdx.x >> 5;
    const int wavesPerBlock = blockDim.x >> 5;
    const int tile = blockIdx.x * wavesPerBlock + wave;
    const int rowBase = tile * 16;
    if (rowBase >= rows) return;                 // uniform per wave

    const int m    = lane & 15;
    const int half = lane >> 4;                  // 0: lanes 0-15, 1: lanes 16-31

    int row = rowBase + m;
    if (row >= rows) row = rows - 1;             // clamp for tail tile; discarded at store
    const float* Wr = W + (size_t)row * (size_t)cols + 4 * half;
    const float* xk = x + 4 * half;

    v8f c0 = {}, c1 = {}, c2 = {}, c3 = {};

    // prologue: preload chunks k=0 and k=8
    v4f a0 = *reinterpret_cast<const v4f*>(Wr);
    v4f b0 = *reinterpret_cast<const v4f*>(xk);
    v4f a1 = *reinterpret_cast<const v4f*>(Wr + 8);
    v4f b1 = *reinterpret_cast<const v4f*>(xk + 8);

    int k = 0;
    for (; k + 16 < cols; k += 16) {
        // issue next-iteration loads before consuming current buffers
        v4f na0 = *reinterpret_cast<const v4f*>(Wr + k + 16);
        v4f nb0 = *reinterpret_cast<const v4f*>(xk + k + 16);
        v4f na1 = *reinterpret_cast<const v4f*>(Wr + k + 24);
        v4f nb1 = *reinterpret_cast<const v4f*>(xk + k + 24);

        v2f aa, bb;
        aa = __builtin_shufflevector(a0, a0, 0, 1);
        bb = __builtin_shufflevector(b0, b0, 0, 1);
        c0 = __builtin_amdgcn_wmma_f32_16x16x4_f32(false, aa, false, bb, (short)0, c0, false, false);
        aa = __builtin_shufflevector(a0, a0, 2, 3);
        bb = __builtin_shufflevector(b0, b0, 2, 3);
        c1 = __builtin_amdgcn_wmma_f32_16x16x4_f32(false, aa, false, bb, (short)0, c1, false, false);
        aa = __builtin_shufflevector(a1, a1, 0, 1);
        bb = __builtin_shufflevector(b1, b1, 0, 1);
        c2 = __builtin_amdgcn_wmma_f32_16x16x4_f32(false, aa, false, bb, (short)0, c2, false, false);
        aa = __builtin_shufflevector(a1, a1, 2, 3);
        bb = __builtin_shufflevector(b1, b1, 2, 3);
        c3 = __builtin_amdgcn_wmma_f32_16x16x4_f32(false, aa, false, bb, (short)0, c3, false, false);

        a0 = na0; b0 = nb0; a1 = na1; b1 = nb1;
    }

    // epilogue: consume last two buffered chunks
    {
        v2f aa, bb;
        aa = __builtin_shufflevector(a0, a0, 0, 1);
        bb = __builtin_shufflevector(b0, b0, 0, 1);
        c0 = __builtin_amdgcn_wmma_f32_16x16x4_f32(false, aa, false, bb, (short)0, c0, false, false);
        aa = __builtin_shufflevector(a0, a0, 2, 3);
        bb = __builtin_shufflevector(b0, b0, 2, 3);
        c1 = __builtin_amdgcn_wmma_f32_16x16x4_f32(false, aa, false, bb, (short)0, c1, false, false);
        aa = __builtin_shufflevector(a1, a1, 0, 1);
        bb = __builtin_shufflevector(b1, b1, 0, 1);
        c2 = __builtin_amdgcn_wmma_f32_16x16x4_f32(false, aa, false, bb, (short)0, c2, false, false);
        aa = __builtin_shufflevector(a1, a1, 2, 3);
        bb = __builtin_shufflevector(b1, b1, 2, 3);
        c3 = __builtin_amdgcn_wmma_f32_16x16x4_f32(false, aa, false, bb, (short)0, c3, false, false);
    }

    union { v8f v; float f[8]; } acc;
    acc.v = (c0 + c1) + (c2 + c3);

    if (lane == 0) {                             // rows rowBase .. rowBase+7
        for (int r = 0; r < 8; ++r) {
            int orow = rowBase + r;
            if (orow < rows) {
                float v = acc.f[r] + bias[orow];
                if (accumulate) v += y[orow];
                y[orow] = v;
            }
        }
    } else if (lane == 16) {                     // rows rowBase+8 .. rowBase+15
        for (int r = 0; r < 8; ++r) {
            int orow = rowBase + 8 + r;
            if (orow < rows) {
                float v = acc.f[r] + bias[orow];
                if (accumulate) v += y[orow];
                y[orow] = v;
            }
        }
    }
}

// attn_in = [ emb[input_index] | h0 ]  (2048)
__global__ void build_attn_in_kernel(const int* __restrict__ input_index,
                                     const float* __restrict__ emb,
                                     const float* __restrict__ h0,
                                     float* __restrict__ attn_in)
{
    int i = blockIdx.x * blockDim.x + threadIdx.x;
    if (i >= EDIM + HDIM) return;
    if (i < EDIM) {
        int idx = input_index[0];
        attn_in[i] = emb[(size_t)idx * EDIM + i];
    } else {
        attn_in[i] = h0[i - EDIM];
    }
}

// in-place softmax over exactly 512 scores; one block of 512 threads
__global__ void softmax512_kernel(float* __restrict__ s)
{
    __shared__ float red[LDIM];
    int tid = threadIdx.x;
    float v = s[tid];
    red[tid] = v; __syncthreads();
    for (int st = LDIM / 2; st > 0; st >>= 1) {
        if (tid < st) red[tid] = fmaxf(red[tid], red[tid + st]);
        __syncthreads();
    }
    float mx = red[0]; __syncthreads();
    float e = expf(v - mx);
    red[tid] = e; __syncthreads();
    for (int st = LDIM / 2; st > 0; st >>= 1) {
        if (tid < st) red[tid] += red[tid + st];
        __syncthreads();
    }
    s[tid] = e / red[0];
}

// attn_applied[h] = sum_l w[l] * enc[l, h]
__global__ void attn_apply_kernel(const float* __restrict__ w,
                                  const float* __restrict__ enc,
                                  float* __restrict__ out)
{
    int h = blockIdx.x * blockDim.x + threadIdx.x;
    if (h >= HDIM) return;
    float acc = 0.0f;
    for (int l = 0; l < LDIM; ++l)
        acc += w[l] * enc[(size_t)l * HDIM + h];
    out[h] = acc;
}

// comb_in = [ embedded | attn_applied ]
__global__ void build_comb_in_kernel(const float* __restrict__ attn_in,
                                     const float* __restrict__ attn_app,
                                     float* __restrict__ comb_in)
{
    int i = blockIdx.x * blockDim.x + threadIdx.x;
    if (i >= EDIM + HDIM) return;
    comb_in[i] = (i < EDIM) ? attn_in[i] : attn_app[i - EDIM];
}

__global__ void tanh_kernel(const float* __restrict__ in, float* __restrict__ out, int n)
{
    int i = blockIdx.x * blockDim.x + threadIdx.x;
    if (i < n) out[i] = tanhf(in[i]);
}

// LSTM cell pointwise (torch gate order i,f,g,o)
__global__ void lstm_cell_kernel(const float* __restrict__ gates,
                                 const float* __restrict__ c0,
                                 float* __restrict__ hnew_ws,
                                 float* __restrict__ out_h,
                                 float* __restrict__ out_c)
{
    int j = blockIdx.x * blockDim.x + threadIdx.x;
    if (j >= HDIM) return;
    float ig = gates[j];
    float fg = gates[HDIM + j];
    float gg = gates[2 * HDIM + j];
    float og = gates[3 * HDIM + j];
    float si = 1.0f / (1.0f + expf(-ig));
    float sf = 1.0f / (1.0f + expf(-fg));
    float so = 1.0f / (1.0f + expf(-og));
    float cn = sf * c0[j] + si * tanhf(gg);
    float hn = so * tanhf(cn);
    hnew_ws[j] = hn;
    out_h[j]   = hn;
    out_c[j]   = cn;
}

// single-block logsumexp over n logits
__global__ void logsumexp_kernel(const float* __restrict__ logits, int n,
                                 float* __restrict__ lse)
{
    __shared__ float red[1024];
    int tid = threadIdx.x;
    float mx = -INFINITY;
    for (int i = tid; i < n; i += blockDim.x) mx = fmaxf(mx, logits[i]);
    red[tid] = mx; __syncthreads();
    for (int st = blockDim.x / 2; st > 0; st >>= 1) {
        if (tid < st) red[tid] = fmaxf(red[tid], red[tid + st]);
        __syncthreads();
    }
    float gm = red[0]; __syncthreads();
    float s = 0.0f;
    for (int i = tid; i < n; i += blockDim.x) s += expf(logits[i] - gm);
    red[tid] = s; __syncthreads();
    for (int st = blockDim.x / 2; st > 0; st >>= 1) {
        if (tid < st) red[tid] += red[tid + st];
        __syncthreads();
    }
    if (tid == 0) *lse = gm + logf(red[0]);
}

__global__ void sub_lse_kernel(float* __restrict__ logits, int n,
                               const float* __restrict__ lse)
{
    int i = blockIdx.x * blockDim.x + threadIdx.x;
    if (i < n) logits[i] -= *lse;
}

extern "C" void kernel_launch(void* const* d_in, const int* in_sizes, int n_in,
                              void* d_out, int out_size, void* d_ws, size_t ws_size,
                              hipStream_t stream)
{
    const int*   input_index = (const int*)  d_in[0];
    const float* h0          = (const float*)d_in[1];
    const float* c0          = (const float*)d_in[2];
    const float* enc         = (const float*)d_in[3];
    const float* emb         = (const float*)d_in[4];
    const float* attn_W      = (const float*)d_in[5];
    const float* attn_b      = (const float*)d_in[6];
    const float* comb_W      = (const float*)d_in[7];
    const float* comb_b      = (const float*)d_in[8];
    const float* W_ih        = (const float*)d_in[9];
    const float* W_hh        = (const float*)d_in[10];
    const float* b_ih        = (const float*)d_in[11];
    const float* b_hh        = (const float*)d_in[12];
    const float* out_W       = (const float*)d_in[13];
    const float* out_b       = (const float*)d_in[14];

    float* out  = (float*)d_out;           // [ log_probs(V) | h_new(H) | c_new(H) ]
    float* ws   = (float*)d_ws;

    float* attn_in  = ws + WS_ATTN_IN;
    float* scores   = ws + WS_SCORES;
    float* attn_app = ws + WS_ATTN_APP;
    float* comb_in  = ws + WS_COMB_IN;
    float* xvec     = ws + WS_XVEC;
    float* gates    = ws + WS_GATES;
    float* hnew     = ws + WS_HNEW;
    float* lse      = ws + WS_LSE;

    // 1. attn_in = [embedded | h]
    build_attn_in_kernel<<<(EDIM + HDIM + 255) / 256, 256, 0, stream>>>(
        input_index, emb, h0, attn_in);

    // 2. scores = attn_W @ attn_in + attn_b   (512 x 2048) -> 32 tiles
    wmma_gemv_kernel<<<4, 256, 0, stream>>>(attn_W, attn_in, attn_b, scores,
                                            LDIM, EDIM + HDIM, 0);

    // 3. softmax over 512 scores (in place)
    softmax512_kernel<<<1, LDIM, 0, stream>>>(scores);

    // 4. attn_applied = weights @ encoder_outputs
    attn_apply_kernel<<<(HDIM + 255) / 256, 256, 0, stream>>>(scores, enc, attn_app);

    // 5. comb_in = [embedded | attn_applied]
    build_comb_in_kernel<<<(EDIM + HDIM + 255) / 256, 256, 0, stream>>>(
        attn_in, attn_app, comb_in);

    // 6. x = tanh(comb_W @ comb_in + comb_b)   (1024 x 2048) -> 64 tiles
    wmma_gemv_kernel<<<8, 256, 0, stream>>>(comb_W, comb_in, comb_b, xvec,
                                            EDIM, EDIM + HDIM, 0);
    tanh_kernel<<<(EDIM + 255) / 256, 256, 0, stream>>>(xvec, xvec, EDIM);

    // 7. gates = W_ih @ x + b_ih + W_hh @ h + b_hh   (4096 x 1024, x2) -> 256 tiles
    wmma_gemv_kernel<<<32, 256, 0, stream>>>(W_ih, xvec, b_ih, gates,
                                             4 * HDIM, EDIM, 0);
    wmma_gemv_kernel<<<32, 256, 0, stream>>>(W_hh, h0, b_hh, gates,
                                             4 * HDIM, HDIM, 1);

    // 8. LSTM cell pointwise; writes h_new/c_new to d_out tail + ws copy of h_new
    lstm_cell_kernel<<<(HDIM + 255) / 256, 256, 0, stream>>>(
        gates, c0, hnew, out + VDIM, out + VDIM + HDIM);

    // 9. logits = out_W @ h_new + out_b   (50257 x 1024) -> 3142 tiles, 8 waves/block
    {
        int tiles  = (VDIM + 15) / 16;
        int blocks = (tiles + 7) / 8;
        wmma_gemv_kernel<<<blocks, 256, 0, stream>>>(out_W, hnew, out_b, out,
                                                     VDIM, HDIM, 0);
    }

    // 10. log_softmax: lse then subtract
    logsumexp_kernel<<<1, 1024, 0, stream>>>(out, VDIM, lse);
    sub_lse_kernel<<<(VDIM + 255) / 256, 256, 0, stream>>>(out, VDIM, lse);
}